// HessBlock_79963701117407
// MI455X (gfx1250) — compile-verified
//
#include <hip/hip_runtime.h>

// Fused: conv3d(5^3, replicate-pad) -> Hessian(6) -> MLP(6->10 relu -> 1 sigmoid)
// MI455X / gfx1250: wave32, WMMA f32 16x16x4 for the per-voxel MLP GEMM.

#define Dv 160
#define TW 16
#define TH 4
#define TD 4
#define XW 24
#define XH 12
#define XD 12
#define YW 20
#define YH 8
#define YD 8
#define GW 18
#define GH 6
#define GD 6

typedef float v2f __attribute__((ext_vector_type(2)));
typedef float v8f __attribute__((ext_vector_type(8)));

// jnp.gradient: central diff interior, one-sided at global edges.
__device__ __forceinline__ float grad1(const float* p, int stride, int g) {
    if (g <= 0)          return p[stride] - p[0];
    else if (g >= Dv-1)  return p[0] - p[-stride];
    else                 return 0.5f * (p[stride] - p[-stride]);
}

__launch_bounds__(256)
__global__ void hessblock_fused(const float* __restrict__ x,
                                const float* __restrict__ conv_w,
                                const float* __restrict__ w1,
                                const float* __restrict__ b1,
                                const float* __restrict__ w2,
                                const float* __restrict__ b2,
                                float* __restrict__ out)
{
    __shared__ float xt[XD*XH*XW];      // 3456 f  (13.8 KB) input tile, halo 4
    __shared__ float yt[YD*YH*YW];      // 1280 f  conv out, halo 2
    __shared__ float gzt[GD*GH*GW];     // 648 f   d/dz, halo 1
    __shared__ float gyt[GD*GH*GW];
    __shared__ float gxt[GD*GH*GW];
    __shared__ float hbuf[8][272];      // 8 feature rows (6 real + 2 zero pads)
    __shared__ float zbuf[16][272];     // layer-1 rows (10 real + 6 pad)
    __shared__ float cw[125];
    __shared__ float w1s[60];
    __shared__ float b1s[16];
    __shared__ float w2s[10];
    __shared__ float b2s;

    const int tid = threadIdx.x;
    const int b   = blockIdx.z & 1;
    const int d0  = (blockIdx.z >> 1) * TD;
    const int h0  = blockIdx.y * TH;
    const int w0  = blockIdx.x * TW;

    // ---- stage 0: constants to LDS ----
    if (tid < 125)           cw[tid]        = conv_w[tid];
    else if (tid < 185)      w1s[tid-125]   = w1[tid-125];
    else if (tid < 195)      b1s[tid-185]   = b1[tid-185];
    else if (tid < 201)      b1s[tid-185]   = 0.f;          // pad rows 10..15
    else if (tid < 211)      w2s[tid-201]   = w2[tid-201];
    else if (tid == 211)     b2s            = b2[0];

    // ---- stage 0b: x tile with replicate (edge) clamp ----
    const size_t xbase = (size_t)b * (Dv*Dv*Dv);
    for (int i = tid; i < XD*XH*XW; i += 256) {
        int lw = i % XW, lh = (i / XW) % XH, ld = i / (XW*XH);
        int gz = min(max(d0 - 4 + ld, 0), Dv-1);
        int gy = min(max(h0 - 4 + lh, 0), Dv-1);
        int gx = min(max(w0 - 4 + lw, 0), Dv-1);
        xt[i] = x[xbase + (size_t)gz*Dv*Dv + (size_t)gy*Dv + gx];
    }
    __syncthreads();

    // ---- stage 1: 5x5x5 conv -> y tile (fp32 VALU, sliding window in W) ----
    // Thread owns 5 consecutive y along W: 8d x 8h x 4 groups = 256 threads.
    // Per (dz,dy): 9 x-loads feed 25 FMAs (vs 25 loads without reuse).
    {
        const int w5 = (tid & 3) * 5;         // 0,5,10,15
        const int h  = (tid >> 2) & 7;
        const int d  = tid >> 5;              // 0..7
        const int base = (d*XH + h)*XW + w5;

        float acc[5] = {0.f, 0.f, 0.f, 0.f, 0.f};
        #pragma unroll
        for (int dz = 0; dz < 5; ++dz) {
            #pragma unroll
            for (int dy = 0; dy < 5; ++dy) {
                const float* xrow = &xt[base + (dz*XH + dy)*XW];
                const float* wrow = &cw[(dz*5 + dy)*5];
                float xr[9];
                #pragma unroll
                for (int j = 0; j < 9; ++j) xr[j] = xrow[j];
                #pragma unroll
                for (int dx = 0; dx < 5; ++dx) {
                    const float wv = wrow[dx];
                    #pragma unroll
                    for (int t = 0; t < 5; ++t)
                        acc[t] = fmaf(wv, xr[t + dx], acc[t]);
                }
            }
        }
        const int ybase = (d*YH + h)*YW + w5;
        #pragma unroll
        for (int t = 0; t < 5; ++t) yt[ybase + t] = acc[t];
    }
    __syncthreads();

    // ---- stage 2: first gradients (halo 1), global-edge one-sided ----
    for (int i = tid; i < GD*GH*GW; i += 256) {
        int w = i % GW, h = (i / GW) % GH, d = i / (GW*GH);
        const float* yc = &yt[((d+1)*YH + (h+1))*YW + (w+1)];
        gzt[i] = grad1(yc, YH*YW, d0 - 1 + d);
        gyt[i] = grad1(yc, YW,    h0 - 1 + h);
        gxt[i] = grad1(yc, 1,     w0 - 1 + w);
    }
    __syncthreads();

    // ---- stage 3: 6 Hessian components for own voxel (+2 zero pad rows) ----
    {
        int w = tid % TW, h = (tid / TW) % TH, d = tid / (TW*TH);
        int gz = d0 + d, gy = h0 + h, gx = w0 + w;
        const int gi = ((d+1)*GH + (h+1))*GW + (w+1);
        const float* pz = &gzt[gi];
        const float* py = &gyt[gi];
        const float* px = &gxt[gi];
        hbuf[0][tid] = grad1(pz, GH*GW, gz);  // d_z g_z
        hbuf[1][tid] = grad1(pz, GW,    gy);  // d_y g_z
        hbuf[2][tid] = grad1(pz, 1,     gx);  // d_x g_z
        hbuf[3][tid] = grad1(py, GW,    gy);  // d_y g_y
        hbuf[4][tid] = grad1(py, 1,     gx);  // d_x g_y
        hbuf[5][tid] = grad1(px, 1,     gx);  // d_x g_x
        hbuf[6][tid] = 0.f;                   // K=6 pad
        hbuf[7][tid] = 0.f;                   // K=7 pad
    }
    __syncthreads();

    // ---- stage 4: MLP layer 1 via V_WMMA_F32_16X16X4_F32 (K=8 in 2 steps) ----
    // D[m,n] = sum_k w1[m,k] * h[k,voxel n] + b1[m]; m<10, k<6 valid.
    {
        const int lane = tid & 31;
        const int wv   = tid >> 5;
        const int half = lane >> 4;     // 0: lanes 0-15, 1: lanes 16-31
        const int lm   = lane & 15;

        // A 16x4 layout: lanes0-15 -> K={0,1}, lanes16-31 -> K={2,3}
        v2f a0, a1;
        {
            int k0 = half ? 2 : 0, k1 = half ? 3 : 1;
            a0.x = (lm < 10) ? w1s[lm*6 + k0] : 0.f;
            a0.y = (lm < 10) ? w1s[lm*6 + k1] : 0.f;
            a1.x = (lm < 10 && k0 + 4 < 6) ? w1s[lm*6 + k0 + 4] : 0.f;
            a1.y = (lm < 10 && k1 + 4 < 6) ? w1s[lm*6 + k1 + 4] : 0.f;
        }

        #pragma unroll
        for (int t = 0; t < 2; ++t) {
            const int vb = wv*32 + t*16;          // 16-voxel N-tile base
            // B 4x16 layout mirrors A: lanes0-15 K={0,1}, lanes16-31 K={2,3}
            v2f bm0, bm1;
            bm0.x = hbuf[half ? 2 : 0][vb + lm];
            bm0.y = hbuf[half ? 3 : 1][vb + lm];
            bm1.x = hbuf[half ? 6 : 4][vb + lm];  // rows 6,7 are zero pads
            bm1.y = hbuf[half ? 7 : 5][vb + lm];
            // C seeded with b1 (rows r / r+8 per half-wave)
            v8f c;
            #pragma unroll
            for (int r = 0; r < 8; ++r) c[r] = b1s[r + (half ? 8 : 0)];
            c = __builtin_amdgcn_wmma_f32_16x16x4_f32(false, a0, false, bm0,
                                                      (short)0, c, false, false);
            c = __builtin_amdgcn_wmma_f32_16x16x4_f32(false, a1, false, bm1,
                                                      (short)0, c, false, false);
            // ReLU via single v_med3_f32; rows 10..15 are dead padding
            #pragma unroll
            for (int r = 0; r < 8; ++r) {
                int m = r + (half ? 8 : 0);
                zbuf[m][vb + lm] = __builtin_amdgcn_fmed3f(c[r], 0.f,
                                                           __builtin_inff());
            }
        }
    }
    __syncthreads();

    // ---- stage 5: layer 2 + sigmoid + coalesced store ----
    {
        float s = b2s;
        #pragma unroll
        for (int m = 0; m < 10; ++m) s = fmaf(w2s[m], zbuf[m][tid], s);
        float r = 1.f / (1.f + __expf(-s));
        int w = tid % TW, h = (tid / TW) % TH, d = tid / (TW*TH);
        out[(size_t)b*(Dv*Dv*Dv) + (size_t)(d0+d)*Dv*Dv + (size_t)(h0+h)*Dv + (w0+w)] = r;
    }
}

extern "C" void kernel_launch(void* const* d_in, const int* in_sizes, int n_in,
                              void* d_out, int out_size, void* d_ws, size_t ws_size,
                              hipStream_t stream) {
    const float* x      = (const float*)d_in[0];
    const float* conv_w = (const float*)d_in[1];
    const float* w1     = (const float*)d_in[2];
    const float* b1     = (const float*)d_in[3];
    const float* w2     = (const float*)d_in[4];
    const float* b2     = (const float*)d_in[5];
    float* out = (float*)d_out;

    dim3 grid(Dv/TW, Dv/TH, (Dv/TD)*2);   // (10, 40, 80): z = dtile*2 + batch
    hessblock_fused<<<grid, 256, 0, stream>>>(x, conv_w, w1, b1, w2, b2, out);
}